// S3FVIN_32220844655299
// MI455X (gfx1250) — compile-verified
//
#include <hip/hip_runtime.h>

// MI455X / gfx1250. Strategy:
//  - Layer-1 of the three 4->10 MLPs via V_WMMA_F32_16X16X4_F32 (K=4 exact fit,
//    bias preloaded into the f32 accumulator), 6 WMMA per wave (2 row-tiles x 3 MLPs).
//  - WMMA C-layout -> per-thread transpose through LDS (stride 50 floats, bank-clean).
//  - Everything else (tanh, layer-2, 3x3 PSD inverse, analytic dual-number Newton
//    Jacobian, quaternion algebra) is scalar per-thread fp32: the workload is
//    VALU/transcendental bound (~2-3 GFLOP vs 64 MB => far above the 23.3 TB/s roofline).

typedef float v2f __attribute__((ext_vector_type(2)));
typedef float v8f __attribute__((ext_vector_type(8)));

#define H_STEP 0.01f
#define EPSF   1e-12f

__device__ __forceinline__ float frcp(float x) { return __builtin_amdgcn_rcpf(x); }

__device__ __forceinline__ float ftanh(float x) {
#if __has_builtin(__builtin_amdgcn_tanhf)
  return __builtin_amdgcn_tanhf(x);            // gfx1250 v_tanh_f32
#else
  // tanh(x) = 1 - 2/(exp(2x)+1), exp(2x) = exp2(2*log2(e)*x)
  float t = __builtin_amdgcn_exp2f(2.8853900817779268f * x);
  return 1.0f - 2.0f * __builtin_amdgcn_rcpf(t + 1.0f);
#endif
}

// o = S*v with S symmetric, packed [S00,S01,S02,S11,S12,S22]
__device__ __forceinline__ void symv(const float S[6], const float v[3], float o[3]) {
  o[0] = S[0]*v[0] + S[1]*v[1] + S[2]*v[2];
  o[1] = S[1]*v[0] + S[3]*v[1] + S[4]*v[2];
  o[2] = S[2]*v[0] + S[4]*v[1] + S[5]*v[2];
}

__device__ __forceinline__ void cross3(const float a[3], const float b[3], float o[3]) {
  o[0] = a[1]*b[2] - a[2]*b[1];
  o[1] = a[2]*b[0] - a[0]*b[2];
  o[2] = a[0]*b[1] - a[1]*b[0];
}

// o = inv(M) * r, general 3x3 (adjugate + v_rcp of det)
__device__ __forceinline__ void solve3(const float M[3][3], const float r[3], float o[3]) {
  float A =  (M[1][1]*M[2][2] - M[1][2]*M[2][1]);
  float Bc = -(M[1][0]*M[2][2] - M[1][2]*M[2][0]);
  float C =  (M[1][0]*M[2][1] - M[1][1]*M[2][0]);
  float det = M[0][0]*A + M[0][1]*Bc + M[0][2]*C;
  float id = frcp(det);
  float D = -(M[0][1]*M[2][2] - M[0][2]*M[2][1]);
  float E =  (M[0][0]*M[2][2] - M[0][2]*M[2][0]);
  float F = -(M[0][0]*M[2][1] - M[0][1]*M[2][0]);
  float G =  (M[0][1]*M[1][2] - M[0][2]*M[1][1]);
  float Hc= -(M[0][0]*M[1][2] - M[0][2]*M[1][0]);
  float I =  (M[0][0]*M[1][1] - M[0][1]*M[1][0]);
  o[0] = (A *r[0] + D *r[1] + G *r[2]) * id;
  o[1] = (Bc*r[0] + E *r[1] + Hc*r[2]) * id;
  o[2] = (C *r[0] + F *r[1] + I *r[2]) * id;
}

__device__ __forceinline__ void quat_mul(const float a[4], const float b[4], float o[4]) {
  o[0] = a[0]*b[0] - a[1]*b[1] - a[2]*b[2] - a[3]*b[3];
  o[1] = a[0]*b[1] + a[1]*b[0] + a[2]*b[3] - a[3]*b[2];
  o[2] = a[0]*b[2] - a[1]*b[3] + a[2]*b[0] + a[3]*b[1];
  o[3] = a[0]*b[3] + a[1]*b[2] - a[2]*b[1] + a[3]*b[0];
}

__global__ __launch_bounds__(256) void s3f_step(
    const float* __restrict__ x,
    const float* __restrict__ Jw1, const float* __restrict__ Jb1,
    const float* __restrict__ Jw2, const float* __restrict__ Jb2,
    const float* __restrict__ Vw1, const float* __restrict__ Vb1,
    const float* __restrict__ Vw2,
    const float* __restrict__ gw1, const float* __restrict__ gb1,
    const float* __restrict__ gw2, const float* __restrict__ gb2,
    float* __restrict__ out)
{
  constexpr int LSTR = 50;                 // floats per LDS row; bank-conflict-free
  __shared__ float hpre[256 * LSTR];       // [thread][ J:0..9 | V:16..25 | g:32..41 ]

  const int tid = threadIdx.x;
  const int lane = tid & 31;
  const int wav  = tid >> 5;
  const int lhi  = lane >> 4;              // half-wave select
  const int llo  = lane & 15;
  const long blockRow0 = (long)blockIdx.x * 256;

  // ---------------- WMMA layer-1 for all three MLPs ----------------
  // B operand (4x16, K x N): lane holds N=llo, VGPR j holds K = 2*lhi + j.
  const int k0 = lhi * 2;
  v2f bJ, bV, bG;
  float cJ, cV, cG;
  if (llo < 10) {
    bJ.x = Jw1[llo*4 + k0]; bJ.y = Jw1[llo*4 + k0 + 1];
    bV.x = Vw1[llo*4 + k0]; bV.y = Vw1[llo*4 + k0 + 1];
    bG.x = gw1[llo*4 + k0]; bG.y = gw1[llo*4 + k0 + 1];
    cJ = Jb1[llo]; cV = Vb1[llo]; cG = gb1[llo];
  } else {
    bJ.x = bJ.y = bV.x = bV.y = bG.x = bG.y = 0.0f;
    cJ = cV = cG = 0.0f;
  }

  #pragma unroll
  for (int t = 0; t < 2; ++t) {
    const int rbase = wav*32 + t*16;                 // local row base of this tile
    const long grow = blockRow0 + rbase + llo;       // A operand: lane = row llo
    v2f a;
    a.x = x[grow*8 + k0];
    a.y = x[grow*8 + k0 + 1];

    v8f accJ, accV, accG;                            // bias replicated over rows
    #pragma unroll
    for (int r = 0; r < 8; ++r) { accJ[r] = cJ; accV[r] = cV; accG[r] = cG; }

    accJ = __builtin_amdgcn_wmma_f32_16x16x4_f32(false, a, false, bJ, (short)0, accJ, false, false);
    accV = __builtin_amdgcn_wmma_f32_16x16x4_f32(false, a, false, bV, (short)0, accV, false, false);
    accG = __builtin_amdgcn_wmma_f32_16x16x4_f32(false, a, false, bG, (short)0, accG, false, false);

    // C layout: VGPR r holds row M = r + 8*lhi, column N = llo.
    const int lrow = rbase + (lhi << 3);
    #pragma unroll
    for (int r = 0; r < 8; ++r) {
      hpre[(lrow + r)*LSTR +      llo] = accJ[r];
      hpre[(lrow + r)*LSTR + 16 + llo] = accV[r];
      hpre[(lrow + r)*LSTR + 32 + llo] = accG[r];
    }
  }
  __syncthreads();

  // ---------------- per-thread scalar path ----------------
  const long row = blockRow0 + tid;
  const float* xr = x + row*8;
  float q[4] = { xr[0], xr[1], xr[2], xr[3] };
  float w[3] = { xr[4], xr[5], xr[6] };
  const float u = xr[7];

  const float* hp = hpre + tid*LSTR;

  // J MLP layer-2 -> l6
  float hJ[10];
  #pragma unroll
  for (int i = 0; i < 10; ++i) hJ[i] = ftanh(hp[i]);
  float l6[6];
  #pragma unroll
  for (int o = 0; o < 6; ++o) {
    float s = Jb2[o];
    #pragma unroll
    for (int i = 0; i < 10; ++i) s += Jw2[o*10 + i]*hJ[i];
    l6[o] = s;
  }
  // g MLP layer-2 -> g3
  float g3[3];
  {
    float hg[10];
    #pragma unroll
    for (int i = 0; i < 10; ++i) hg[i] = ftanh(hp[32 + i]);
    #pragma unroll
    for (int o = 0; o < 3; ++o) {
      float s = gb2[o];
      #pragma unroll
      for (int i = 0; i < 10; ++i) s += gw2[o*10 + i]*hg[i];
      g3[o] = s;
    }
  }
  // dV at qk from cached V pre-activations: dV = W1^T (w2 .* (1-h^2))
  float dV[4] = {0,0,0,0};
  #pragma unroll
  for (int i = 0; i < 10; ++i) {
    float h = ftanh(hp[16 + i]);
    float s = Vw2[i]*(1.0f - h*h);
    #pragma unroll
    for (int j = 0; j < 4; ++j) dV[j] += s*Vw1[i*4 + j];
  }

  // J_q_inv = L L^T + 0.01 I (symmetric), L lower-tri from l6
  float Ai[6];
  Ai[0] = l6[0]*l6[0] + 0.01f;
  Ai[1] = l6[0]*l6[1];
  Ai[2] = l6[0]*l6[3];
  Ai[3] = l6[1]*l6[1] + l6[2]*l6[2] + 0.01f;
  Ai[4] = l6[1]*l6[3] + l6[2]*l6[4];
  Ai[5] = l6[3]*l6[3] + l6[4]*l6[4] + l6[5]*l6[5] + 0.01f;

  // J_q = inv(J_q_inv) (symmetric adjugate)
  float Jq[6];
  {
    float c00 = Ai[3]*Ai[5] - Ai[4]*Ai[4];
    float c01 = Ai[2]*Ai[4] - Ai[1]*Ai[5];
    float c02 = Ai[1]*Ai[4] - Ai[2]*Ai[3];
    float c11 = Ai[0]*Ai[5] - Ai[2]*Ai[2];
    float c12 = Ai[1]*Ai[2] - Ai[0]*Ai[4];
    float c22 = Ai[0]*Ai[3] - Ai[1]*Ai[1];
    float id = frcp(Ai[0]*c00 + Ai[1]*c01 + Ai[2]*c02);
    Jq[0]=c00*id; Jq[1]=c01*id; Jq[2]=c02*id; Jq[3]=c11*id; Jq[4]=c12*id; Jq[5]=c22*id;
  }

  float fk[3] = { 0.005f*g3[0]*u, 0.005f*g3[1]*u, 0.005f*g3[2]*u };  // h*g*u/2

  float pk[3];
  symv(Jq, w, pk);
  pk[0] *= 2.0f; pk[1] *= 2.0f; pk[2] *= 2.0f;

  // Hmat(qk) @ dV = qv*dV0 + q0*dVv - qv x dVv
  float rhs[3];
  {
    float qv[3] = { q[1], q[2], q[3] };
    float dVv[3] = { dV[1], dV[2], dV[3] };
    float cx[3]; cross3(qv, dVv, cx);
    #pragma unroll
    for (int i = 0; i < 3; ++i) {
      float Hdv = qv[i]*dV[0] + q[0]*dVv[i] - cx[i];
      rhs[i] = -0.25f*H_STEP*(pk[i] + fk[i] - 0.5f*H_STEP*Hdv);
    }
  }

  // ---------------- Newton (4 iters), analytic forward-mode Jacobian ----------------
  float xi[3] = {0,0,0};
  #pragma unroll
  for (int it = 0; it < 4; ++it) {
    float s = xi[0]*xi[0] + xi[1]*xi[1] + xi[2]*xi[2];
    float th = sqrtf(fmaxf(s, EPSF));
    float rth = frcp(th);
    float sn = sinf(th), cs = cosf(th);
    float sinc = sn*rth;
    float qv[3] = { -xi[0]*sinc, -xi[1]*sinc, -xi[2]*sinc };
    float m[3]; symv(Jq, qv, m);
    float cqm[3]; cross3(qv, m, cqm);
    float res[3] = { cs*m[0]-cqm[0]-rhs[0], cs*m[1]-cqm[1]-rhs[1], cs*m[2]-cqm[2]-rhs[2] };

    float dth_s   = (s > EPSF) ? rth : 0.0f;   // matches jax max(s,EPS) clamp
    float dsinc_s = (cs - sinc)*rth;
    float Jm[3][3];
    #pragma unroll
    for (int k = 0; k < 3; ++k) {
      float dth = xi[k]*dth_s;
      float dc  = -sn*dth;
      float dsc = dsinc_s*dth;
      float dqv[3];
      #pragma unroll
      for (int i = 0; i < 3; ++i) dqv[i] = -xi[i]*dsc - ((i == k) ? sinc : 0.0f);
      float dm[3]; symv(Jq, dqv, dm);
      float t1[3]; cross3(dqv, m, t1);
      float t2[3]; cross3(qv, dm, t2);
      #pragma unroll
      for (int i = 0; i < 3; ++i)
        Jm[i][k] = dc*m[i] + cs*dm[i] - t1[i] - t2[i];
    }
    float dx[3];
    solve3(Jm, res, dx);
    xi[0] -= dx[0]; xi[1] -= dx[1]; xi[2] -= dx[2];
  }

  // qk_next = qk * exp(xi)
  float qn[4];
  {
    float s = xi[0]*xi[0] + xi[1]*xi[1] + xi[2]*xi[2];
    float th = sqrtf(fmaxf(s, EPSF));
    float sinc = sinf(th)*frcp(th);
    float e[4] = { cosf(th), xi[0]*sinc, xi[1]*sinc, xi[2]*sinc };
    quat_mul(q, e, qn);
  }

  // dV at qk_next (full scalar forward + backprop of V MLP)
  float dVn[4] = {0,0,0,0};
  #pragma unroll
  for (int i = 0; i < 10; ++i) {
    float p = Vb1[i] + Vw1[i*4+0]*qn[0] + Vw1[i*4+1]*qn[1]
                     + Vw1[i*4+2]*qn[2] + Vw1[i*4+3]*qn[3];
    float h = ftanh(p);
    float sc = Vw2[i]*(1.0f - h*h);
    #pragma unroll
    for (int j = 0; j < 4; ++j) dVn[j] += sc*Vw1[i*4 + j];
  }

  // qq = conj(qk) * qk_next ; pk_next ; wk_next
  float qc[4] = { q[0], -q[1], -q[2], -q[3] };
  float qq[4]; quat_mul(qc, qn, qq);
  float qqv[3] = { qq[1], qq[2], qq[3] };
  float mm[3]; symv(Jq, qqv, mm);
  float cq[3]; cross3(qqv, mm, cq);

  float qnv[3]  = { qn[1], qn[2], qn[3] };
  float dVnv[3] = { dVn[1], dVn[2], dVn[3] };
  float cn[3]; cross3(qnv, dVnv, cn);

  float pn[3];
  #pragma unroll
  for (int i = 0; i < 3; ++i) {
    float Gq = qq[0]*mm[i] - cq[i];                       // Gmat(qq) @ (Jq qq_v)
    float Hd = qnv[i]*dVn[0] + qn[0]*dVnv[i] - cn[i];     // Hmat(qn) @ dVn
    pn[i] = 400.0f*Gq - 0.005f*Hd + fk[i];                // 4/h = 400, h/2 = 0.005
  }
  float wn[3]; symv(Ai, pn, wn);                          // 0.5 * J_q_inv @ pn
  wn[0] *= 0.5f; wn[1] *= 0.5f; wn[2] *= 0.5f;

  float* orow = out + row*8;
  orow[0] = qn[0]; orow[1] = qn[1]; orow[2] = qn[2]; orow[3] = qn[3];
  orow[4] = wn[0]; orow[5] = wn[1]; orow[6] = wn[2]; orow[7] = u;
}

extern "C" void kernel_launch(void* const* d_in, const int* in_sizes, int n_in,
                              void* d_out, int out_size, void* d_ws, size_t ws_size,
                              hipStream_t stream) {
  const float* x   = (const float*)d_in[0];
  const float* Jw1 = (const float*)d_in[1];
  const float* Jb1 = (const float*)d_in[2];
  const float* Jw2 = (const float*)d_in[3];
  const float* Jb2 = (const float*)d_in[4];
  const float* Vw1 = (const float*)d_in[5];
  const float* Vb1 = (const float*)d_in[6];
  const float* Vw2 = (const float*)d_in[7];
  // d_in[8] = Vb2: unused (only grad of V is needed)
  const float* gw1 = (const float*)d_in[9];
  const float* gb1 = (const float*)d_in[10];
  const float* gw2 = (const float*)d_in[11];
  const float* gb2 = (const float*)d_in[12];
  float* out = (float*)d_out;

  const int B = in_sizes[0] / 8;          // 1048576; divisible by 256
  const int blocks = B / 256;
  s3f_step<<<blocks, 256, 0, stream>>>(x, Jw1, Jb1, Jw2, Jb2,
                                       Vw1, Vb1, Vw2,
                                       gw1, gb1, gw2, gb2, out);
}